// CausalSelfAttention_55009941127944
// MI455X (gfx1250) — compile-verified
//
#include <hip/hip_runtime.h>
#include <hip/hip_bf16.h>

typedef __attribute__((ext_vector_type(2))) float v2f;
typedef __attribute__((ext_vector_type(8))) float v8f;

#define B_  2
#define T_  2048
#define C_  1024
#define H_  16
#define HD  64

__device__ __forceinline__ v8f wmma_f32(v2f a, v2f b, v8f c) {
    // D = A(16x4) * B(4x16) + C(16x16), fp32, wave32
    return __builtin_amdgcn_wmma_f32_16x16x4_f32(
        false, a, false, b, (short)0, c, false, false);
}

// ---------------------------------------------------------------------------
// GEMM: out[M x N] = A[M x K] * W[K x N] + bias.  One wave computes 16x64.
// mode 0: plain store to out.  mode 1: scatter q/k/v -> [B,H,T,hd] buffers.
// ---------------------------------------------------------------------------
__global__ void __launch_bounds__(32)
gemm16x64_kernel(const float* __restrict__ A, const float* __restrict__ W,
                 const float* __restrict__ bias, int K, int N, int mode,
                 float* __restrict__ out, float* __restrict__ qb,
                 float* __restrict__ kb, float* __restrict__ vb) {
    const int lane = threadIdx.x & 31;
    const int half = lane >> 4;     // 0: K rows 0,1   1: K rows 2,3
    const int ln   = lane & 15;
    const int row0 = blockIdx.y * 16;
    const int col0 = blockIdx.x * 64;

    v8f acc[4];
#pragma unroll
    for (int nt = 0; nt < 4; ++nt)
#pragma unroll
        for (int i = 0; i < 8; ++i) acc[nt][i] = 0.0f;

    const float* arow = A + (size_t)(row0 + ln) * K + 2 * half;

    for (int k0 = 0; k0 < K; k0 += 4) {
        v2f afrag = *(const v2f*)(arow + k0);
        const float* wrow = W + (size_t)(k0 + 2 * half) * N + col0 + ln;
        __builtin_prefetch(arow + k0 + 64, 0, 0);   // global_prefetch_b8
#pragma unroll
        for (int nt = 0; nt < 4; ++nt) {
            v2f bfrag;
            bfrag.x = wrow[nt * 16];        // B[k0+2*half  ][col]
            bfrag.y = wrow[(size_t)N + nt * 16]; // B[k0+2*half+1][col]
            acc[nt] = wmma_f32(afrag, bfrag, acc[nt]);
        }
    }

#pragma unroll
    for (int nt = 0; nt < 4; ++nt) {
#pragma unroll
        for (int i = 0; i < 8; ++i) {
            const int r = row0 + i + 8 * half;
            const int c = col0 + nt * 16 + ln;
            const float val = acc[nt][i] + bias[c];
            if (mode == 0) {
                out[(size_t)r * N + c] = val;
            } else {
                // c in [0,3C): section (q/k/v), head, dim
                const int sec = c >> 10;          // /C_
                const int cc  = c & (C_ - 1);
                const int h   = cc >> 6;          // /HD
                const int d   = cc & (HD - 1);
                const int b   = r >> 11;          // /T_
                const int t   = r & (T_ - 1);
                float* dst = (sec == 0) ? qb : (sec == 1) ? kb : vb;
                dst[((size_t)(b * H_ + h) * T_ + t) * HD + d] = val;
            }
        }
    }
}

// ---------------------------------------------------------------------------
// RoPE in place on q and k laid out [B,H,T,hd].  One thread per (.., pair d).
// ---------------------------------------------------------------------------
__global__ void rope_kernel(float* __restrict__ q, float* __restrict__ k) {
    const int j = blockIdx.x * blockDim.x + threadIdx.x;
    if (j >= B_ * H_ * T_ * (HD / 2)) return;
    const int d = j & 31;
    const int t = (j >> 5) & (T_ - 1);
    const size_t base = (size_t)(j >> 5) * HD;
    // inv_freq = 10000^(-d/32) = exp(-d/32 * ln 10000)
    const float ang = (float)t * __expf(-(float)d * (9.210340371976184f / 32.0f));
    float sn, cs;
    __sincosf(ang, &sn, &cs);
    const float q1 = q[base + d], q2 = q[base + d + 32];
    q[base + d]      = q1 * cs - q2 * sn;
    q[base + d + 32] = q2 * cs + q1 * sn;
    const float k1 = k[base + d], k2 = k[base + d + 32];
    k[base + d]      = k1 * cs - k2 * sn;
    k[base + d + 32] = k2 * cs + k1 * sn;
}

// ---------------------------------------------------------------------------
// Flash attention: one wave per (b,h, 16-query tile).  Online softmax.
// q,k,v: [B,H,T,hd].  y: [B,T,C] (heads re-interleaved for the proj GEMM).
// ---------------------------------------------------------------------------
__global__ void __launch_bounds__(32)
attn_kernel(const float* __restrict__ q, const float* __restrict__ k,
            const float* __restrict__ v, float* __restrict__ y) {
    __shared__ float pshm[16 * 16];
    const int lane = threadIdx.x & 31;
    const int half = lane >> 4;
    const int ln   = lane & 15;
    const int bh   = blockIdx.y;        // b*H + h
    const int b    = bh >> 4;
    const int h    = bh & 15;
    const int qt   = blockIdx.x;
    const int q0   = qt * 16;

    const float* Q = q + (size_t)bh * T_ * HD;
    const float* K = k + (size_t)bh * T_ * HD;
    const float* V = v + (size_t)bh * T_ * HD;

    // Load Q fragments (A layout), fold in softmax scale 1/sqrt(hd).
    v2f qa[16];
    {
        const float* qrow = Q + (size_t)(q0 + ln) * HD + 2 * half;
#pragma unroll
        for (int c = 0; c < 16; ++c) {
            v2f t = *(const v2f*)(qrow + c * 4);
            qa[c].x = t.x * 0.125f;
            qa[c].y = t.y * 0.125f;
        }
    }

    float m[8], l[8];
    v8f o[4];
#pragma unroll
    for (int i = 0; i < 8; ++i) { m[i] = -3.0e38f; l[i] = 0.0f; }
#pragma unroll
    for (int nt = 0; nt < 4; ++nt)
#pragma unroll
        for (int i = 0; i < 8; ++i) o[nt][i] = 0.0f;

    for (int j = 0; j <= qt; ++j) {
        const int k0 = j * 16;

        // S = Q * K^T  (B fragment: B[kd][n] = K[k0+n][kd])
        v8f s;
#pragma unroll
        for (int i = 0; i < 8; ++i) s[i] = 0.0f;
        const float* krow = K + (size_t)(k0 + ln) * HD + 2 * half;
#pragma unroll
        for (int c = 0; c < 16; ++c) {
            v2f kf = *(const v2f*)(krow + c * 4);
            s = wmma_f32(qa[c], kf, s);
        }

        // Causal mask + online softmax (16-lane groups match C/D layout).
        float corr[8];
#pragma unroll
        for (int i = 0; i < 8; ++i) {
            const int row = q0 + i + 8 * half;
            const int col = k0 + ln;
            float sv = (col <= row) ? s[i] : -3.0e38f;
            float rm = sv;
            rm = fmaxf(rm, __shfl_xor(rm, 1, 32));
            rm = fmaxf(rm, __shfl_xor(rm, 2, 32));
            rm = fmaxf(rm, __shfl_xor(rm, 4, 32));
            rm = fmaxf(rm, __shfl_xor(rm, 8, 32));
            const float mnew = fmaxf(m[i], rm);
            const float p = __expf(sv - mnew);
            float rs = p;
            rs += __shfl_xor(rs, 1, 32);
            rs += __shfl_xor(rs, 2, 32);
            rs += __shfl_xor(rs, 4, 32);
            rs += __shfl_xor(rs, 8, 32);
            corr[i] = __expf(m[i] - mnew);
            l[i] = l[i] * corr[i] + rs;
            m[i] = mnew;
            pshm[(i + 8 * half) * 16 + ln] = p;   // P in row-major LDS tile
        }
        __syncthreads();

        // Re-read P as A fragments (layout conversion via LDS).
        v2f pa[4];
#pragma unroll
        for (int c = 0; c < 4; ++c)
            pa[c] = *(const v2f*)(&pshm[ln * 16 + c * 4 + 2 * half]);

        // O = diag(corr)*O + P*V
#pragma unroll
        for (int nt = 0; nt < 4; ++nt) {
#pragma unroll
            for (int i = 0; i < 8; ++i) o[nt][i] *= corr[i];
#pragma unroll
            for (int c = 0; c < 4; ++c) {
                const float* vp = V + (size_t)(k0 + c * 4 + 2 * half) * HD + nt * 16 + ln;
                v2f vf;
                vf.x = vp[0];
                vf.y = vp[HD];
                o[nt] = wmma_f32(pa[c], vf, o[nt]);
            }
        }
        __syncthreads();
    }

    // Normalize and store into [B,T,C] with head h at columns h*hd..h*hd+63.
    float* yrow = y + ((size_t)b * T_ + q0) * C_ + h * HD;
#pragma unroll
    for (int nt = 0; nt < 4; ++nt)
#pragma unroll
        for (int i = 0; i < 8; ++i) {
            const int row = i + 8 * half;
            yrow[(size_t)row * C_ + nt * 16 + ln] = o[nt][i] / l[i];
        }
}

// ---------------------------------------------------------------------------
extern "C" void kernel_launch(void* const* d_in, const int* in_sizes, int n_in,
                              void* d_out, int out_size, void* d_ws, size_t ws_size,
                              hipStream_t stream) {
    (void)in_sizes; (void)n_in; (void)out_size; (void)ws_size;
    const float* x      = (const float*)d_in[0];
    const float* W_attn = (const float*)d_in[1];
    const float* b_attn = (const float*)d_in[2];
    const float* W_proj = (const float*)d_in[3];
    const float* b_proj = (const float*)d_in[4];
    float* out = (float*)d_out;

    const size_t per = (size_t)B_ * H_ * T_ * HD;   // 4,194,304 floats
    float* qb = (float*)d_ws;
    float* kb = qb + per;
    float* vb = kb + per;
    float* yb = vb + per;                           // [B,T,C]

    // 1) QKV GEMM + bias, scatter into per-head buffers
    dim3 g1((3 * C_) / 64, (B_ * T_) / 16);
    gemm16x64_kernel<<<g1, 32, 0, stream>>>(x, W_attn, b_attn, C_, 3 * C_, 1,
                                            nullptr, qb, kb, vb);
    // 2) RoPE on q and k
    const int nrope = B_ * H_ * T_ * (HD / 2);
    rope_kernel<<<(nrope + 255) / 256, 256, 0, stream>>>(qb, kb);
    // 3) Causal flash attention
    dim3 g2(T_ / 16, B_ * H_);
    attn_kernel<<<g2, 32, 0, stream>>>(qb, kb, vb, yb);
    // 4) Output projection GEMM + bias
    dim3 g3(C_ / 64, (B_ * T_) / 16);
    gemm16x64_kernel<<<g3, 32, 0, stream>>>(yb, W_proj, b_proj, C_, C_, 0,
                                            out, nullptr, nullptr, nullptr);
}